// ModelNew_50388556316951
// MI455X (gfx1250) — compile-verified
//
#include <hip/hip_runtime.h>

// CDNA5 / gfx1250: fp32 WMMA 16x16x4 transposed-conv via parity decomposition.
// A and B fragments are loaded as contiguous 64-bit (float2) quantities so they
// land directly in even-aligned VGPR pairs (no repack movs).
typedef __attribute__((ext_vector_type(2))) float v2f;
typedef __attribute__((ext_vector_type(8))) float v8f;

#define LDS_ROW 136                 // ix-line length in float2 units; 4*136 % 64 == 32 -> halves on disjoint banks
#define LDS_PAIR_STRIDE (2 * LDS_ROW * 2)  // floats per cin-pair (2 rows interleaved x2)

// Packed weights (in d_ws): wp[tap][cpair][cout][2]; wp[t][p][c] = (w[2p,c,t], w[2p+1,c,t])
__global__ __launch_bounds__(256)
void ModelNew_wpack_kernel(const float* __restrict__ w, float* __restrict__ wp) {
  const int idx = blockIdx.x * 256 + threadIdx.x;  // over 9*32*64 = 18432 float2s
  if (idx >= 9 * 32 * 64) return;
  const int t = idx >> 11;           // / 2048
  const int rem = idx & 2047;
  const int p = rem >> 6;
  const int c = rem & 63;
  wp[2 * idx    ] = w[(2 * p    ) * 576 + c * 9 + t];
  wp[2 * idx + 1] = w[(2 * p + 1) * 576 + c * 9 + t];
}

// One parity class (py,px): D[cout, sx] += sum_cin sum_tap W[cin,cout,tap] * X[cin, row(tap), sx+sh(tap)]
// A (16x4, MxK): VGPR v, lane-half h -> K = v + 2h, M = lane%16  => cins (k0+2h, k0+2h+1) = pair (k0>>1)+h
// B (4x16, KxN): same K mapping, N = lane%16                     => same cin pair, one ds_load_b64
// C/D (16x16):   VGPR i, half h -> M = i + 8h, N = lane%16
template <int NTAP>
__device__ __forceinline__ void compute_class(
    const float* __restrict__ wp, const float* __restrict__ bias,
    float* __restrict__ out, const float* xlds,
    int n, int sy, int cout0, int py, int px,
    const int rowt[NTAP], const int ixsht[NTAP], const int wit[NTAP]) {
  const int lane = threadIdx.x & 31;
  const int h    = lane >> 4;   // lane half
  const int nn   = lane & 15;   // N (spatial) / M (cout) index within half

  v8f acc[8];
#pragma unroll
  for (int t = 0; t < 8; ++t) {
#pragma unroll
    for (int i = 0; i < 8; ++i) acc[t][i] = 0.0f;
  }

  const int aoff = (cout0 + nn) * 2;  // lane's cout slot in packed weights (float2 granularity *2 floats)

  for (int k0 = 0; k0 < 64; k0 += 4) {
    const int p = (k0 >> 1) + h;      // cin-pair index for this lane half
    v2f afr[NTAP];
#pragma unroll
    for (int t = 0; t < NTAP; ++t) {  // one coalesced b64 per tap, reused by all 8 spatial tiles
      afr[t] = *(const v2f*)(&wp[(wit[t] * 32 + p) * 128 + aoff]);
    }
#pragma unroll
    for (int tile = 0; tile < 8; ++tile) {
      const int sxn = tile * 16 + nn;
#pragma unroll
      for (int t = 0; t < NTAP; ++t) {
        // interleaved pair layout: one aligned b64 fetches (cin, cin+1) at (row, ix)
        const v2f bfr = *(const v2f*)(&xlds[((p * 2 + rowt[t]) * LDS_ROW + sxn + ixsht[t]) * 2]);
        acc[tile] = __builtin_amdgcn_wmma_f32_16x16x4_f32(
            false, afr[t], false, bfr, (short)0, acc[tile], false, false);
      }
    }
  }

  // Epilogue: bias + fused clamp(0, LIMIT=0.5), strided NCHW store.
  const int oy = 2 * sy + py;
  float bv[8];
#pragma unroll
  for (int i = 0; i < 8; ++i) bv[i] = bias[cout0 + i + 8 * h];
#pragma unroll
  for (int tile = 0; tile < 8; ++tile) {
    const int ox = 2 * (tile * 16 + nn) + px;
#pragma unroll
    for (int i = 0; i < 8; ++i) {
      const int cout = cout0 + i + 8 * h;
      float v = acc[tile][i] + bv[i];
      v = fminf(fmaxf(v, 0.0f), 0.5f);
      out[(((size_t)n * 64 + cout) * 256 + oy) * 256 + ox] = v;
    }
  }
}

__global__ __launch_bounds__(256)
void ModelNew_convt2x_wmma_kernel(const float* __restrict__ x,
                                  const float* __restrict__ wp,
                                  const float* __restrict__ b,
                                  float* __restrict__ out) {
  // xl[cpair][row][ix][2]: rows sy & sy+1 for all 32 cin-pairs, cins interleaved. 68 KB.
  __shared__ float xlds[32 * LDS_PAIR_STRIDE];

  const int n   = blockIdx.x >> 7;
  const int sy  = blockIdx.x & 127;
  const int tid = threadIdx.x;
  const bool row1valid = (sy + 1) < 128;

  // Cooperative fill: 32 pairs x 2 rows x 33 float4 segments (cols 0..131; col>=128 zeroed).
  for (int idx = tid; idx < 32 * 2 * 33; idx += 256) {
    const int cp  = idx / 66;
    const int rem = idx - cp * 66;
    const int row = rem / 33;
    const int seg = rem - row * 33;
    const int ix  = seg * 4;
    float4 a = make_float4(0.0f, 0.0f, 0.0f, 0.0f);
    float4 c = a;
    if (ix < 128 && (row == 0 || row1valid)) {
      const size_t base = (((size_t)n * 64 + 2 * cp) * 128 + (sy + row)) * 128 + ix;
      a = *(const float4*)(x + base);          // cin = 2*cp
      c = *(const float4*)(x + base + 16384);  // cin = 2*cp+1  (128*128 apart)
    }
    float4 lo = make_float4(a.x, c.x, a.y, c.y);
    float4 hi = make_float4(a.z, c.z, a.w, c.w);
    float* dst = &xlds[((cp * 2 + row) * LDS_ROW + ix) * 2];
    *(float4*)(dst)     = lo;
    *(float4*)(dst + 4) = hi;
  }
  __syncthreads();

  const int wave  = tid >> 5;          // 8 waves
  const int cout0 = (wave & 3) << 4;   // 4 cout tiles of 16

  // Parity classes: taps ky == (oy+1)&1 set, kx == (ox+1)&1 set.
  // row: iy==sy+1 -> 1, iy==sy -> 0; ixsh: kx==0 -> +1, kx==2 -> 0; widx = ky*3+kx.
  if ((wave >> 2) == 0) {  // classes (0,0) [1 tap] + (1,1) [4 taps]  => 5 taps
    {
      const int rows[1] = {0}, shs[1] = {0}, wis[1] = {4};
      compute_class<1>(wp, b, out, xlds, n, sy, cout0, 0, 0, rows, shs, wis);
    }
    {
      const int rows[4] = {1, 1, 0, 0}, shs[4] = {1, 0, 1, 0}, wis[4] = {0, 2, 6, 8};
      compute_class<4>(wp, b, out, xlds, n, sy, cout0, 1, 1, rows, shs, wis);
    }
  } else {                 // classes (0,1) [2 taps] + (1,0) [2 taps] => 4 taps
    {
      const int rows[2] = {0, 0}, shs[2] = {1, 0}, wis[2] = {3, 5};
      compute_class<2>(wp, b, out, xlds, n, sy, cout0, 0, 1, rows, shs, wis);
    }
    {
      const int rows[2] = {1, 0}, shs[2] = {0, 0}, wis[2] = {1, 7};
      compute_class<2>(wp, b, out, xlds, n, sy, cout0, 1, 0, rows, shs, wis);
    }
  }
}

extern "C" void kernel_launch(void* const* d_in, const int* in_sizes, int n_in,
                              void* d_out, int out_size, void* d_ws, size_t ws_size,
                              hipStream_t stream) {
  const float* x = (const float*)d_in[0];  // (32, 64, 128, 128)
  const float* w = (const float*)d_in[1];  // (64, 64, 3, 3)  [cin, cout, ky, kx]
  const float* b = (const float*)d_in[2];  // (64,)
  float* out = (float*)d_out;              // (32, 64, 256, 256)
  float* wp  = (float*)d_ws;               // packed weights: 9*32*64 float2 = 147,456 B

  // 1) repack weights for b64 fragment loads (deterministic, runs every call)
  hipLaunchKernelGGL(ModelNew_wpack_kernel, dim3(72), dim3(256), 0, stream, w, wp);
  // 2) main WMMA kernel: one block per (image, sub-row sy)
  hipLaunchKernelGGL(ModelNew_convt2x_wmma_kernel, dim3(32 * 128), dim3(256), 0, stream,
                     x, wp, b, out);
}